// TemporalSampling_60266981097691
// MI455X (gfx1250) — compile-verified
//
#include <hip/hip_runtime.h>
#include <math.h>

typedef float v2f __attribute__((ext_vector_type(2)));
typedef float v8f __attribute__((ext_vector_type(8)));

#define NBATCH 16
#define CCH    256
#define T      64
#define NB     64
#define KDIM   512
#define HW     196     // 14*14
#define NTILES 13      // ceil(196/16)
#define EPS    1e-8f

// ---------------------------------------------------------------------------
// Phase 1: build the Gaussian resampling matrix kernels[n][b][t]  (16x64x64)
// One block per n, 256 threads.
// ---------------------------------------------------------------------------
__global__ void __launch_bounds__(256)
ts_build_kernels(const float* __restrict__ emb, float* __restrict__ kern) {
    __shared__ float s_mean[KDIM];
    __shared__ float s_dist[T];
    __shared__ float s_red[256];
    __shared__ float s_scal[3];   // [0]=normM, [1]=scale, [2]=step

    const int n   = blockIdx.x;
    const int tid = threadIdx.x;
    const float* E = emb + (size_t)n * KDIM * T;

    // mean over t for each k (2 rows per thread), plus ||mean||^2 partial
    float m2 = 0.0f;
    for (int r = 0; r < 2; ++r) {
        const int k = tid * 2 + r;
        float s = 0.0f;
        for (int t = 0; t < T; ++t) s += E[k * T + t];
        const float mk = s * (1.0f / (float)T);
        s_mean[k] = mk;
        m2 = fmaf(mk, mk, m2);
    }
    s_red[tid] = m2;
    __syncthreads();
    for (int off = 128; off > 0; off >>= 1) {
        if (tid < off) s_red[tid] += s_red[tid + off];
        __syncthreads();
    }
    if (tid == 0) s_scal[0] = sqrtf(s_red[0]);
    __syncthreads();
    const float normM = s_scal[0];

    // cosine distance per frame t
    if (tid < T) {
        float num = 0.0f, e2 = 0.0f;
        for (int k = 0; k < KDIM; ++k) {
            const float v = E[k * T + tid];       // coalesced across tid
            num = fmaf(v, s_mean[k], num);
            e2  = fmaf(v, v, e2);
        }
        const float den = sqrtf(e2) * normM + EPS;
        s_dist[tid] = 1.0f - num / den;
    }
    __syncthreads();
    if (tid == 0) {
        float mx = s_dist[0];
        for (int t = 1; t < T; ++t) mx = fmaxf(mx, s_dist[t]);
        s_scal[1] = (2.0f * (float)NB) / mx;
    }
    __syncthreads();
    if (tid < T) s_dist[tid] *= s_scal[1];
    __syncthreads();
    if (tid == 0) {
        float mx = s_dist[0];
        for (int t = 1; t < T; ++t) mx = fmaxf(mx, s_dist[t]);
        s_scal[2] = mx / (float)NB;               // bin step
    }
    __syncthreads();
    const float step     = s_scal[2];
    const float inv_step = 1.0f / step;

    // 256 threads cover the 64x64 kernel matrix: thread -> (b, 16 t's)
    const int b  = tid >> 2;
    const int t0 = (tid & 3) * 16;
    const float mu = fmaf(step, (float)b, step * 0.5f);
    float* Kout = kern + ((size_t)n * NB + b) * T;
    for (int t = t0; t < t0 + 16; ++t) {
        const float z = (s_dist[t] - mu) * inv_step;
        Kout[t] = expf(-0.5f * z * z);
    }
}

// ---------------------------------------------------------------------------
// Phase 2: out[n][c][b][hw] = sum_t kern[n][b][t] * inp[n][c][t][hw]
// One block per (c, n): 128 threads = 4 waves; wave w owns bin rows 16w..16w+15.
// [64x64] x [64x196] GEMM via V_WMMA_F32_16X16X4_F32.
// X tile staged global->LDS with GLOBAL_LOAD_ASYNC_TO_LDS_B128 (ASYNCcnt),
// bypassing VGPRs entirely.
// ---------------------------------------------------------------------------
typedef __attribute__((address_space(3))) float lds_float_t;

__global__ void __launch_bounds__(128)
ts_resample(const float* __restrict__ inp, const float* __restrict__ kern,
            float* __restrict__ out, float* __restrict__ binsum) {
    __shared__ float Xs[T * HW + 16];   // 64x196 fp32 tile (+pad for N-tile overread)

    const int c    = blockIdx.x;
    const int n    = blockIdx.y;
    const int tid  = threadIdx.x;
    const int wave = tid >> 5;
    const int lane = tid & 31;
    const int hi16 = lane >> 4;     // lane half (0: lanes 0-15, 1: lanes 16-31)
    const int l15  = lane & 15;

    const float* X = inp + (size_t)(n * CCH + c) * (T * HW);

#if defined(__gfx1250__)
    // Async stage: 12544 floats = 3136 x b128, one 16B chunk per active lane
    // per issue; data flows memory -> LDS without touching VGPRs.
    {
        const unsigned lds0 = (unsigned)(uintptr_t)(lds_float_t*)&Xs[0];
        for (int i = tid; i < (T * HW) / 4; i += 128) {
            const unsigned           ldsa = lds0 + (unsigned)i * 16u;
            const unsigned long long ga   = (unsigned long long)(const void*)(X + (size_t)i * 4);
            asm volatile("global_load_async_to_lds_b128 %0, %1, off"
                         :: "v"(ldsa), "v"(ga) : "memory");
        }
    }
#else
    // Fallback path (host parse / non-gfx1250): copy through VGPRs.
    {
        const float4* Xv = (const float4*)X;
        float4*       Sv = (float4*)Xs;
        for (int i = tid; i < (T * HW) / 4; i += 128) Sv[i] = Xv[i];
    }
#endif

    // Preload A fragments (overlaps with the async staging): wave's 16 bin
    // rows, all 16 k-steps. A layout (16x4 f32): lanes 0-15 hold K={0,1},
    // lanes 16-31 hold K={2,3}.
    v2f a[16];
    {
        const float* Kb = kern + ((size_t)n * NB + (wave * 16 + l15)) * T + hi16 * 2;
        const v2f*   Kv = (const v2f*)Kb;   // 8B-aligned (even float index)
#pragma unroll
        for (int s = 0; s < 16; ++s) a[s] = Kv[2 * s];
    }

#if defined(__gfx1250__)
    asm volatile("s_wait_asynccnt 0x0" ::: "memory");
#endif
    __syncthreads();

    float bs[8];
#pragma unroll
    for (int r = 0; r < 8; ++r) bs[r] = 0.0f;

    for (int nt = 0; nt < NTILES; ++nt) {
        const int  col = nt * 16 + l15;
        const bool cv  = (col < HW);

        v8f acc = {};
#pragma unroll
        for (int s = 0; s < 16; ++s) {
            // B layout (4x16 f32): VGPR j holds K=j (lanes 0-15), K=j+2 (16-31).
            const int r0 = 4 * s + hi16 * 2;
            v2f b;
            b.x = Xs[r0 * HW + col];
            b.y = Xs[(r0 + 1) * HW + col];
            acc = __builtin_amdgcn_wmma_f32_16x16x4_f32(
                /*neg_a=*/false, a[s], /*neg_b=*/false, b,
                /*c_mod=*/(short)0, acc, /*reuse_a=*/false, /*reuse_b=*/false);
        }

        // D layout: VGPR r -> bin row (r + 8*lanehalf), column = lane%16.
        // Output is written once and never re-read: non-temporal stores keep
        // the 192MB L2 free for the hot kernel matrix.
        float* Ob = out + ((size_t)(n * CCH + c) * NB + (wave * 16 + hi16 * 8)) * HW + col;
        if (cv) {
#pragma unroll
            for (int r = 0; r < 8; ++r) {
                __builtin_nontemporal_store(acc[r], &Ob[r * HW]);
                bs[r] += acc[r];
            }
        }
    }

    // Reduce per-bin partial sums across the 16 lanes of each half, then
    // accumulate globally for the mask. XOR masks 1..8 stay inside a half.
#pragma unroll
    for (int r = 0; r < 8; ++r) {
        float v = bs[r];
        v += __shfl_xor(v, 1);
        v += __shfl_xor(v, 2);
        v += __shfl_xor(v, 4);
        v += __shfl_xor(v, 8);
        if (l15 == 0) {
            const int b = wave * 16 + hi16 * 8 + r;
            atomicAdd(&binsum[n * NB + b], v);
        }
    }
}

// ---------------------------------------------------------------------------
// Phase 3: mask[n][b] = (binsum != 0)
// ---------------------------------------------------------------------------
__global__ void ts_mask(const float* __restrict__ binsum, float* __restrict__ mask) {
    const int i = blockIdx.x * blockDim.x + threadIdx.x;
    if (i < NBATCH * NB) mask[i] = (binsum[i] != 0.0f) ? 1.0f : 0.0f;
}

// ---------------------------------------------------------------------------
extern "C" void kernel_launch(void* const* d_in, const int* in_sizes, int n_in,
                              void* d_out, int out_size, void* d_ws, size_t ws_size,
                              hipStream_t stream) {
    const float* inp = (const float*)d_in[0];   // [16,256,64,14,14]
    const float* emb = (const float*)d_in[1];   // [16,512,64]
    float* out  = (float*)d_out;                // [16,256,64,14,14] then mask [16,64]
    float* kern   = (float*)d_ws;                       // 16*64*64 floats
    float* binsum = kern + (size_t)NBATCH * NB * T;     // 16*64 floats
    float* mask   = out + (size_t)NBATCH * CCH * NB * HW;

    hipMemsetAsync(binsum, 0, (size_t)NBATCH * NB * sizeof(float), stream);

    ts_build_kernels<<<NBATCH, 256, 0, stream>>>(emb, kern);

    dim3 grid(CCH, NBATCH);
    ts_resample<<<grid, 128, 0, stream>>>(inp, kern, out, binsum);

    ts_mask<<<(NBATCH * NB + 255) / 256, 256, 0, stream>>>(binsum, mask);
}